// ParametricPart_74818330296517
// MI455X (gfx1250) — compile-verified
//
#include <hip/hip_runtime.h>
#include <stdint.h>

// Burgers-type 3-tap stencil over [ROWS, 1024] f32, memory-bound (~23 us HBM floor).
// CDNA5 path: TDM tensor_load_to_lds DMAs an 8x1024 f32 tile into LDS (TENSORcnt),
// wave 0 waits s_wait_tensorcnt then split-barrier hands off to 8 wave32 consumers.
// Consumers read one ds_load_b128 + 2 halo scalars per row and emit NT b128 stores.

#define D          1024
#define TILE_ROWS  8
#define BLOCK      256

typedef unsigned int u32x4 __attribute__((ext_vector_type(4)));
typedef int          i32x8 __attribute__((ext_vector_type(8)));
typedef int          i32x4 __attribute__((ext_vector_type(4)));
typedef float        f32x4 __attribute__((ext_vector_type(4)));

typedef __attribute__((address_space(3))) float lds_float;

__global__ __launch_bounds__(BLOCK)
void burgers_stencil_tdm_kernel(const float* __restrict__ state,
                                const float* __restrict__ cdp,
                                const float* __restrict__ nvp,
                                float* __restrict__ out,
                                int rows)
{
    __shared__ float lds[TILE_ROWS][D];   // 32 KB tile -> up to 10 blocks/WGP vs 320 KB

    const int tid  = threadIdx.x;
    const int row0 = blockIdx.x * TILE_ROWS;
    const int trows = (rows - row0) < TILE_ROWS ? (rows - row0) : TILE_ROWS;

    // ---- wave 0 issues the TDM DMA: global[row0 .. row0+trows) -> LDS ----
    if (tid < 32) {
        const uint64_t ga    = (uint64_t)(uintptr_t)(state + (size_t)row0 * D);
        const uint32_t lbase = (uint32_t)(size_t)(lds_float*)(&lds[0][0]);

        // D# group 0: count=1 | lds_addr | global_addr[56:0] | type=2
        u32x4 g0;
        g0[0] = 0x1u;                                            // count=1, user desc
        g0[1] = lbase;                                           // lds_addr (bytes)
        g0[2] = (uint32_t)ga;                                    // global_addr[31:0]
        g0[3] = (uint32_t)((ga >> 32) & 0x01FFFFFFu)             // global_addr[56:32]
              | (2u << 30);                                      // type = 2 ("image")

        // D# group 1: data_size=4B, 2-D tile: tile_dim0=1024, tile_dim1=trows,
        // tensor_dim0=1024, tensor_dim0_stride=1024. Exact tiling -> OOB never hits.
        const uint32_t td1 = 0x7FFFFFFFu;
        i32x8 g1;
        g1[0] = (int)(2u << 16);                                 // data_size=2 (4 bytes)
        g1[1] = (int)(((uint32_t)D & 0xFFFFu) << 16);            // tensor_dim0[15:0]
        g1[2] = (int)(((uint32_t)D >> 16)                        // tensor_dim0[31:16]=0
              |       ((td1 & 0xFFFFu) << 16));                  // tensor_dim1[15:0]
        g1[3] = (int)((td1 >> 16)                                // tensor_dim1[31:16]
              |       ((uint32_t)D << 16));                      // tile_dim0 = 1024
        g1[4] = (int)(uint32_t)trows;                            // tile_dim1, tile_dim2=0
        g1[5] = (int)D;                                          // tensor_dim0_stride[31:0]
        g1[6] = 0;                                               // stride hi / dim1_stride lo
        g1[7] = 0;                                               // dim1_stride hi

        i32x4 gz = {0, 0, 0, 0};                                 // groups 2/3 unused (2-D)
#if __clang_major__ >= 23
        i32x8 gz8 = {0, 0, 0, 0, 0, 0, 0, 0};
        __builtin_amdgcn_tensor_load_to_lds(g0, g1, gz, gz, gz8, 0);
#else
        __builtin_amdgcn_tensor_load_to_lds(g0, g1, gz, gz, 0);
#endif
        __builtin_amdgcn_s_wait_tensorcnt(0);                    // DMA landed in LDS
    }
    __syncthreads();                                             // release consumer waves

    const float cd      = *cdp;
    const float nv      = *nvp;
    const float dx      = 0.2f;
    const float inv_dx2 = 1.0f / (dx * dx);
    const float inv_2dx = 1.0f / (2.0f * dx);
    const float ccoef   = -cd * 0.5f * 1.0f * 0.5f;              // -C_D*0.5*H_V*B_V

    const int col0 = tid * 4;                                    // 256 thr * 4 = 1024 cols
    const bool isL = (col0 == 0);                                // lane 0 of wave 0
    const bool isR = (col0 == D - 4);                            // lane 31 of wave 7
    const int  li  = isL ? 0       : col0 - 1;                   // in-bounds halo indices
    const int  ri  = isR ? (D - 1) : col0 + 4;

    for (int r = 0; r < TILE_ROWS; ++r) {
        if (row0 + r >= rows) break;
        const float* Lr = &lds[r][0];

        // window {e0..e5} = Lr[col0-1 .. col0+4]: 1x ds_load_b128 + 2 scalars
        const f32x4 v  = *reinterpret_cast<const f32x4*>(&Lr[col0]);
        const float e0 = Lr[li];
        const float e1 = v.x, e2 = v.y, e3 = v.z, e4 = v.w;
        const float e5 = Lr[ri];

        // edge fixup (copy neighbor column) == shift stencil window by one:
        // k=0 at col 0    uses (e1,e2,e3) instead of (e0,e1,e2)
        // k=3 at col 1023 uses (e2,e3,e4) instead of (e3,e4,e5)
        const float sm0 = isL ? e1 : e0, sc0 = isL ? e2 : e1, sp0 = isL ? e3 : e2;
        const float sm3 = isR ? e2 : e3, sc3 = isR ? e3 : e4, sp3 = isR ? e4 : e5;

        f32x4 res;
        {   // k = 0
            const float uxx = (sp0 - 2.0f * sc0 + sm0) * inv_dx2;
            const float ux  = (sp0 - sm0) * inv_2dx;
            res.x = fmaf(ccoef * fabsf(e1), e1, fmaf(nv, uxx, -e1 * ux));
        }
        {   // k = 1 (always interior)
            const float uxx = (e3 - 2.0f * e2 + e1) * inv_dx2;
            const float ux  = (e3 - e1) * inv_2dx;
            res.y = fmaf(ccoef * fabsf(e2), e2, fmaf(nv, uxx, -e2 * ux));
        }
        {   // k = 2 (always interior)
            const float uxx = (e4 - 2.0f * e3 + e2) * inv_dx2;
            const float ux  = (e4 - e2) * inv_2dx;
            res.z = fmaf(ccoef * fabsf(e3), e3, fmaf(nv, uxx, -e3 * ux));
        }
        {   // k = 3
            const float uxx = (sp3 - 2.0f * sc3 + sm3) * inv_dx2;
            const float ux  = (sp3 - sm3) * inv_2dx;
            res.w = fmaf(ccoef * fabsf(e4), e4, fmaf(nv, uxx, -e4 * ux));
        }

        // streaming output, no reuse -> non-temporal b128 store
        f32x4* dst = reinterpret_cast<f32x4*>(out + (size_t)(row0 + r) * D + col0);
        __builtin_nontemporal_store(res, dst);
    }
}

extern "C" void kernel_launch(void* const* d_in, const int* in_sizes, int n_in,
                              void* d_out, int out_size, void* d_ws, size_t ws_size,
                              hipStream_t stream) {
    const float* state = (const float*)d_in[0];
    const float* cdp   = (const float*)d_in[1];
    const float* nvp   = (const float*)d_in[2];
    float*       out   = (float*)d_out;

    const int rows = in_sizes[0] / D;                     // 65536
    const int grid = (rows + TILE_ROWS - 1) / TILE_ROWS;  // 8192 blocks

    burgers_stencil_tdm_kernel<<<grid, BLOCK, 0, stream>>>(state, cdp, nvp, out, rows);
}